// TophatIntegrator_88673894793500
// MI455X (gfx1250) — compile-verified
//
#include <hip/hip_runtime.h>

// Causal boxcar mean, window 513, T = 2^24 fp32.
// out[t] = (1/513) * sum_{k=0..512} x[t-k], x[<0] = 0.
//
// Single-pass, block-local algorithm (no global scan; the anchor cancels):
//   block b owns outputs [b*CHUNK, (b+1)*CHUNK)
//   stage x[b*CHUNK-528 .. b*CHUNK+CHUNK) into LDS  (async-to-LDS when possible)
//   lp[i]  = inclusive prefix within each 16-element group  (WMMA f32 16x16x4)
//   gpx[g] = exclusive prefix of group sums                  (block scan)
//   out    = (gpx[i/16]+lp[i] - gpx[j/16]-lp[j]) / 513,  j = i-513

#define WINDOW    513
#define CHUNK     8192
#define NTHREADS  256
#define PADL      528                 // 513 halo rounded up to 16-multiple
#define EXT_REAL  (CHUNK + PADL)      // 8720 elements staged from global
#define EXT_PAD   8960                // 35 tiles of 256 (zero-padded tail)
#define NTILES    (EXT_PAD / 256)     // 35
#define NGROUPS   (EXT_PAD / 16)      // 560

typedef float v2f __attribute__((ext_vector_type(2)));
typedef float v4f __attribute__((ext_vector_type(4)));
typedef float v8f __attribute__((ext_vector_type(8)));

// Async-to-LDS builtin pointee type per hipcc diagnostic:
//   'int __attribute__((__vector_size__(4 * sizeof(int)))) __device__ *'
typedef int v4i_vs __attribute__((vector_size(16)));
typedef __attribute__((address_space(1))) v4i_vs* async_gptr;
typedef __attribute__((address_space(3))) v4i_vs* async_lptr;

#if defined(__has_builtin)
#if __has_builtin(__builtin_amdgcn_global_load_async_to_lds_b128) && \
    __has_builtin(__builtin_amdgcn_s_wait_asynccnt)
#define USE_ASYNC 1
#endif
#endif
#ifndef USE_ASYNC
#define USE_ASYNC 0
#endif

__global__ __launch_bounds__(NTHREADS)
void tophat_boxcar_kernel(const float* __restrict__ x,
                          float* __restrict__ out,
                          int n) {
  __shared__ float s_ext[EXT_PAD];     // raw x, then overwritten with lp[]
  __shared__ float s_gpx[NGROUPS];     // exclusive prefix of group sums
  __shared__ float s_ts[NTHREADS];     // block-scan temp

  const int  tid    = threadIdx.x;
  const long bstart = (long)blockIdx.x * CHUNK;
  const long S      = bstart - PADL;   // global index of s_ext[0]

  // ---- Phase 1: stage extended chunk into LDS ----
  const bool interior = (S >= 0) && (S + EXT_REAL <= (long)n);
#if USE_ASYNC
  if (interior) {
    // GLOBAL_LOAD_ASYNC_TO_LDS_B128: 16B/lane direct to LDS, ASYNCcnt-tracked.
    for (int v = tid; v < EXT_REAL / 4; v += NTHREADS) {
      __builtin_amdgcn_global_load_async_to_lds_b128(
          (async_gptr)(x + S + 4L * v),
          (async_lptr)(s_ext + 4 * v),
          0, 0);
    }
    // zero-fill padded tail [EXT_REAL, EXT_PAD) with plain ds stores
    for (int v = EXT_REAL / 4 + tid; v < EXT_PAD / 4; v += NTHREADS) {
      v4f z = {0.f, 0.f, 0.f, 0.f};
      *(v4f*)(s_ext + 4 * v) = z;
    }
    __builtin_amdgcn_s_wait_asynccnt(0);   // wave's async LDS writes complete
  } else
#endif
  {
    for (int v = tid; v < EXT_PAD / 4; v += NTHREADS) {
      v4f val = {0.f, 0.f, 0.f, 0.f};
      if (v < EXT_REAL / 4) {
        long g = S + 4L * v;
        if (g >= 0 && g + 3 < (long)n) {
          val = *(const v4f*)(x + g);          // global_load_b128
        } else {                                // block 0 / ragged tail only
          #pragma unroll
          for (int c = 0; c < 4; ++c) {
            long j = g + c;
            val[c] = (j >= 0 && j < (long)n) ? x[j] : 0.f;
          }
        }
      }
      *(v4f*)(s_ext + 4 * v) = val;
    }
  }
  __syncthreads();

  // ---- Phase 2: 16-element segmented inclusive prefixes via WMMA ----
  // Tile = 16x16 fp32 matrix A, A[m][k] = s_ext[tb + 16m + k].
  // D = A x U, U[k][n] = (k <= n)  ==> D[m][n] = prefix of row m up to n.
  // Four chained V_WMMA_F32_16X16X4_F32 accumulate over K in slabs of 4.
  {
    const int wid  = tid >> 5;        // wave id (uniform per wave)
    const int lane = tid & 31;
    const int row  = lane & 15;       // A row / B,D column (ISA 7.12.2)
    const int half = lane >> 4;       // selects K pair within a slab
    for (int tile = wid; tile < NTILES; tile += (NTHREADS / 32)) {
      const int tb = tile * 256;
      v8f acc = {0.f, 0.f, 0.f, 0.f, 0.f, 0.f, 0.f, 0.f};
      #pragma unroll
      for (int s = 0; s < 4; ++s) {
        const int kb = 4 * s + 2 * half;
        // A slab 16x4: lane holds A[row][kb], A[row][kb+1]
        v2f a = *(const v2f*)(s_ext + tb + 16 * row + kb);
        // B slab 4x16 of upper-triangular ones: lane holds U[kb][col], U[kb+1][col]
        v2f bm;
        bm.x = (kb     <= row) ? 1.0f : 0.0f;
        bm.y = (kb + 1 <= row) ? 1.0f : 0.0f;
        acc = __builtin_amdgcn_wmma_f32_16x16x4_f32(
            false, a, false, bm, (short)0, acc, false, false);
      }
      // D layout: reg r, lane -> element (r + 8*half, col); write lp over ext
      #pragma unroll
      for (int r = 0; r < 8; ++r)
        s_ext[tb + 16 * r + 128 * half + row] = acc[r];
    }
  }
  __syncthreads();

  // ---- Phase 3: exclusive scan of the 560 group sums (CS[g] = lp[16g+15]) ----
  {
    const int g0 = 3 * tid;
    float s0 = (g0     < NGROUPS) ? s_ext[16 * g0       + 15] : 0.f;
    float s1 = (g0 + 1 < NGROUPS) ? s_ext[16 * (g0 + 1) + 15] : 0.f;
    float s2 = (g0 + 2 < NGROUPS) ? s_ext[16 * (g0 + 2) + 15] : 0.f;
    float tot = s0 + s1 + s2;
    s_ts[tid] = tot;
    __syncthreads();
    #pragma unroll
    for (int d = 1; d < NTHREADS; d <<= 1) {   // Hillis-Steele, 8 rounds
      float vv = (tid >= d) ? s_ts[tid - d] : 0.f;
      __syncthreads();
      s_ts[tid] += vv;
      __syncthreads();
    }
    float base = s_ts[tid] - tot;              // exclusive thread offset
    if (g0     < NGROUPS) s_gpx[g0]     = base;
    if (g0 + 1 < NGROUPS) s_gpx[g0 + 1] = base + s0;
    if (g0 + 2 < NGROUPS) s_gpx[g0 + 2] = base + s0 + s1;
  }
  __syncthreads();

  // ---- Phase 4: out[t] = (pre(i) - pre(i-513)) / 513, coalesced b128 stores ----
  const float inv = 1.0f / (float)WINDOW;
  #pragma unroll
  for (int k = 0; k < CHUNK / (4 * NTHREADS); ++k) {   // 8 iterations
    const int tl = k * (4 * NTHREADS) + tid * 4;       // 0..8188 step 4
    const int i  = PADL + tl;                          // ext index (16B aligned)
    v4f lpp = *(const v4f*)(s_ext + i);                // ds_load_b128
    const float gpl = s_gpx[i >> 4];                   // i..i+3 share a group
    v4f o;
    #pragma unroll
    for (int c = 0; c < 4; ++c) {
      const int j = i - WINDOW + c;                    // j >= 15 always
      float pm = s_gpx[j >> 4] + s_ext[j];
      o[c] = (gpl + lpp[c] - pm) * inv;
    }
    long t = bstart + tl;
    if (t + 3 < (long)n) {
      *(v4f*)(out + t) = o;                            // global_store_b128
    } else {
      #pragma unroll
      for (int c = 0; c < 4; ++c)
        if (t + c < (long)n) out[t + c] = o[c];
    }
  }
}

extern "C" void kernel_launch(void* const* d_in, const int* in_sizes, int n_in,
                              void* d_out, int out_size, void* d_ws, size_t ws_size,
                              hipStream_t stream) {
  const float* x = (const float*)d_in[0];
  float* out = (float*)d_out;
  const int n = in_sizes[0];                  // 16,777,216
  const int grid = (n + CHUNK - 1) / CHUNK;   // 2048 blocks, 8 waves each
  tophat_boxcar_kernel<<<grid, NTHREADS, 0, stream>>>(x, out, n);
}